// CapsuleLayer_10557029614085
// MI455X (gfx1250) — compile-verified
//
#include <hip/hip_runtime.h>

// CapsuleLayer dynamic routing, fused 3-pass recompute strategy for MI455X.
// W (134MB) is re-read each pass and stays L2-resident (192MB L2); priors are
// never materialized in HBM. Per-node GEMM uses V_WMMA_F32_16X16X4_F32, and
// each node's 64KB weight slab is DMA'd into LDS by the Tensor Data Mover
// (tensor_load_to_lds, double-buffered, synced with s_wait_tensorcnt).

#define BDIM  32          // batch
#define NNODE 2048        // nodes
#define KCAPS 32          // capsules
#define ICH   16          // in channels
#define OCH   32          // out channels
#define COLS  (KCAPS * OCH)       // 1024 columns of the per-node GEMM
#define WSZ   (KCAPS * ICH * OCH) // 16384 floats = 64KB per node
#define NPG   8           // nodes per workgroup
#define BLOCK 256         // 8 wave32

typedef __attribute__((ext_vector_type(2))) float    v2f;
typedef __attribute__((ext_vector_type(8))) float    v8f;
typedef __attribute__((ext_vector_type(4))) unsigned u32x4;
typedef __attribute__((ext_vector_type(8))) unsigned u32x8;

__device__ __forceinline__ v8f wmma_f32(v2f a, v2f b, v8f c) {
  // D = A(16x4 f32) x B(4x16 f32) + C(16x16 f32)
  return __builtin_amdgcn_wmma_f32_16x16x4_f32(
      /*neg_a=*/false, a, /*neg_b=*/false, b,
      /*c_mod=*/(short)0, c, /*reuse_a=*/false, /*reuse_b=*/false);
}

// TDM: DMA one 64KB weight slab (16384 f32, contiguous) from global to LDS.
// D# group0: count=1 | lds_addr | global_addr[56:0] | type=2
// D# group1: data_size=4B, tensor_dim0=tile_dim0=16384, dim1=1, stride0=16384
__device__ __forceinline__ void tdm_load_wn(const float* gsrc, float* lds_dst) {
  const unsigned lds_off = (unsigned)(unsigned long long)(void*)lds_dst;
  const unsigned long long ga = (unsigned long long)gsrc;
  u32x4 g0;
  g0[0] = 1u;                                                  // count=1
  g0[1] = lds_off;                                             // lds_addr
  g0[2] = (unsigned)ga;                                        // gaddr[31:0]
  g0[3] = (unsigned)((ga >> 32) & 0x01FFFFFFull) | 0x80000000u;// gaddr[56:32]|type=2
  u32x8 g1;
  g1[0] = 0x00020000u;   // workgroup_mask=0, data_size=2 (4B)
  g1[1] = 0x40000000u;   // tensor_dim0[15:0]=16384 in bits[31:16]
  g1[2] = 0x00010000u;   // tensor_dim0[31:16]=0, tensor_dim1=1
  g1[3] = 0x40000000u;   // tile_dim0=16384 in bits[31:16]
  g1[4] = 0x00000001u;   // tile_dim1=1, tile_dim2=0
  g1[5] = 0x00004000u;   // tensor_dim0_stride=16384
  g1[6] = 0u;
  g1[7] = 0u;
  const u32x4 gz = {0u, 0u, 0u, 0u};  // groups 2/3 unused (tile_dim2==0)
  asm volatile("tensor_load_to_lds %0, %1, %2, %3"
               :: "s"(g0), "s"(g1), "s"(gz), "s"(gz) : "memory");
}

// One pass of routing: TDM-stage W_n, recompute priors per node via WMMA,
// (iter>0) compute delta->logits->softmax(probs), then accumulate
// s += probs*priors directly in the WMMA accumulators by row-scaling the A
// fragments. Flush once with f32 atomics.
template <int ITER>
__global__ __launch_bounds__(BLOCK) void caps_pass(
    const float* __restrict__ x,        // [B, N, IC]
    const float* __restrict__ W,        // [N, K, IC, OC]
    const float* __restrict__ outs_prev,// [B, K, OC] (ITER>0)
    float* __restrict__ logits,         // [B, N, K]
    float* __restrict__ s_glob)         // [B, K, OC] accumulated via atomics
{
  __shared__ float sh_w[2][WSZ];                           // 2 x 64 KB (TDM)
  __shared__ float sh_x[BDIM * ICH];                       // 2 KB
  __shared__ float sh_pl[BDIM * KCAPS];                    // 4 KB logits/probs
  __shared__ float sh_pri[(ITER > 0) ? (BDIM * COLS) : 1]; // 128 KB priors

  const int tid   = threadIdx.x;
  const int wave  = tid >> 5;        // wave32
  const int lane  = tid & 31;
  const int h     = wave & 1;        // batch half: b in [h*16, h*16+16)
  const int tbase = (wave >> 1) * 16;// 16 column-tiles per wave
  const int lrow  = lane & 15;       // M row / N col within tile
  const int lhi   = lane >> 4;       // K-pair selector per WMMA layout

  const v8f vz = {0.f, 0.f, 0.f, 0.f, 0.f, 0.f, 0.f, 0.f};
  v8f acc[16];
  #pragma unroll
  for (int j = 0; j < 16; ++j) acc[j] = vz;

  const int n0 = blockIdx.x * NPG;

  // Prefetch first node's weights via TDM (one instruction, wave 0).
  if (wave == 0) tdm_load_wn(W + (size_t)n0 * WSZ, &sh_w[0][0]);

  for (int nn = 0; nn < NPG; ++nn) {
    const int n   = n0 + nn;
    const int buf = nn & 1;

    __syncthreads();  // previous node done reading sh_x / sh_pl / sh_w[buf^1]

    for (int j = tid; j < BDIM * ICH; j += BLOCK) {
      const int b = j >> 4, i = j & 15;
      sh_x[j] = x[(b * NNODE + n) * ICH + i];
    }
    if (wave == 0) {
      __builtin_amdgcn_s_wait_tensorcnt(0);       // sh_w[buf] DMA complete
      if (nn + 1 < NPG)                           // prefetch next node
        tdm_load_wn(W + (size_t)(n + 1) * WSZ, &sh_w[buf ^ 1][0]);
    }
    __syncthreads();  // publish sh_x and sh_w[buf]

    // Raw A fragments: lanes 0-15 hold K = 4c+{0,1}, lanes 16-31 K = 4c+{2,3}
    v2f a[4];
    #pragma unroll
    for (int c = 0; c < 4; ++c) {
      const int i0 = 4 * c + lhi * 2;
      const int bb = h * 16 + lrow;
      a[c].x = sh_x[bb * ICH + i0];
      a[c].y = sh_x[bb * ICH + i0 + 1];
    }

    const float* wb = &sh_w[buf][0];

    if (ITER > 0) {
      // ---- raw priors tile [32 x 1024] into LDS via WMMA (B from LDS) ----
      #pragma unroll
      for (int j = 0; j < 16; ++j) {
        const int t  = tbase + j;
        const int k  = t >> 1;
        const int o0 = (t & 1) * 16;
        v8f d = vz;
        #pragma unroll
        for (int c = 0; c < 4; ++c) {
          const int i0 = 4 * c + lhi * 2;
          const float* wp = wb + (k * ICH + i0) * OCH + o0 + lrow;
          v2f bf; bf.x = wp[0]; bf.y = wp[OCH];
          d = wmma_f32(a[c], bf, d);
        }
        const int colg = t * 16 + lrow;
        #pragma unroll
        for (int r = 0; r < 8; ++r)
          sh_pri[(h * 16 + r + 8 * lhi) * COLS + colg] = d[r];
      }
      __syncthreads();

      // ---- delta = <priors, outputs_prev>_o ; logits update ----
      for (int p = tid; p < BDIM * KCAPS; p += BLOCK) {
        const int b = p & 31;          // lane-major in b to spread LDS banks
        const int k = p >> 5;
        float dsum = 0.f;
        #pragma unroll
        for (int o = 0; o < OCH; ++o) {
          const int oo = (o + b) & 31; // rotate to avoid bank conflicts
          dsum += sh_pri[b * COLS + k * OCH + oo] *
                  outs_prev[b * COLS + k * OCH + oo];
        }
        const int li = (b * NNODE + n) * KCAPS + k;
        float lg = dsum;
        if (ITER == 2) lg += logits[li];
        logits[li] = lg;
        sh_pl[b * KCAPS + k] = lg;
      }
      __syncthreads();

      // ---- softmax over k per b (32 threads) ----
      if (tid < BDIM) {
        float m = -1e30f;
        #pragma unroll
        for (int k = 0; k < KCAPS; ++k) m = fmaxf(m, sh_pl[tid * KCAPS + k]);
        float ssum = 0.f;
        #pragma unroll
        for (int k = 0; k < KCAPS; ++k) {
          const float e = __expf(sh_pl[tid * KCAPS + k] - m);
          sh_pl[tid * KCAPS + k] = e;
          ssum += e;
        }
        const float inv = 1.f / ssum;
        #pragma unroll
        for (int k = 0; k < KCAPS; ++k) sh_pl[tid * KCAPS + k] *= inv;
      }
      __syncthreads();
    }

    // ---- s accumulation: scale A rows by probs[b,n,k], WMMA-accumulate ----
    #pragma unroll
    for (int j = 0; j < 16; ++j) {
      const int t  = tbase + j;
      const int k  = t >> 1;
      const int o0 = (t & 1) * 16;
      float scale = 1.f;
      if (ITER > 0) scale = sh_pl[(h * 16 + lrow) * KCAPS + k];
      #pragma unroll
      for (int c = 0; c < 4; ++c) {
        const int i0 = 4 * c + lhi * 2;
        v2f as; as.x = a[c].x * scale; as.y = a[c].y * scale;
        const float* wp = wb + (k * ICH + i0) * OCH + o0 + lrow;
        v2f bf; bf.x = wp[0]; bf.y = wp[OCH];
        acc[j] = wmma_f32(as, bf, acc[j]);
      }
    }
  }

  // ---- flush partial s to global with f32 atomics ----
  const float fin = (ITER == 0) ? (1.f / 32.f) : 1.f;  // uniform softmax(0)
  #pragma unroll
  for (int j = 0; j < 16; ++j) {
    const int t = tbase + j;
    const int colg = t * 16 + lrow;
    #pragma unroll
    for (int r = 0; r < 8; ++r)
      atomicAdd(&s_glob[(h * 16 + r + 8 * lhi) * COLS + colg], acc[j][r] * fin);
  }
}

__global__ void zero_f32(float* __restrict__ p, int n) {
  const int i = blockIdx.x * blockDim.x + threadIdx.x;
  if (i < n) p[i] = 0.f;
}

__global__ void squash_k(const float* __restrict__ s, float* __restrict__ out) {
  const int row = blockIdx.x * blockDim.x + threadIdx.x;  // (b,k) row
  if (row >= BDIM * KCAPS) return;
  const float* v = s + row * OCH;
  float sq = 0.f;
  #pragma unroll
  for (int o = 0; o < OCH; ++o) sq += v[o] * v[o];
  const float scale = (sq / (1.f + sq)) * rsqrtf(sq);
  #pragma unroll
  for (int o = 0; o < OCH; ++o) out[row * OCH + o] = v[o] * scale;
}

extern "C" void kernel_launch(void* const* d_in, const int* in_sizes, int n_in,
                              void* d_out, int out_size, void* d_ws, size_t ws_size,
                              hipStream_t stream) {
  const float* x = (const float*)d_in[0];        // [32, 2048, 16]
  const float* W = (const float*)d_in[1];        // [2048, 32, 16, 32]
  float* dout    = (float*)d_out;                // [32,1,32,1,32] = 32768 f32

  const int SN = BDIM * KCAPS * OCH;             // 32768
  float* s      = (float*)d_ws;                  // [B,K,O]
  float* outs   = s + SN;                        // [B,K,O]
  float* logits = outs + SN;                     // [B,N,K] = 2M floats

  const dim3 gp(NNODE / NPG), bp(BLOCK);
  const dim3 gz((SN + 255) / 256), bz(256);
  const dim3 gq((BDIM * KCAPS + 127) / 128), bq(128);

  // Iteration 0: uniform probs (1/32)
  zero_f32<<<gz, bz, 0, stream>>>(s, SN);
  caps_pass<0><<<gp, bp, 0, stream>>>(x, W, outs, logits, s);
  squash_k<<<gq, bq, 0, stream>>>(s, outs);

  // Iteration 1: logits = delta(outputs0); probs = softmax(logits)
  zero_f32<<<gz, bz, 0, stream>>>(s, SN);
  caps_pass<1><<<gp, bp, 0, stream>>>(x, W, outs, logits, s);
  squash_k<<<gq, bq, 0, stream>>>(s, outs);

  // Iteration 2: logits += delta(outputs1); final squash to d_out
  zero_f32<<<gz, bz, 0, stream>>>(s, SN);
  caps_pass<2><<<gp, bp, 0, stream>>>(x, W, outs, logits, s);
  squash_k<<<gq, bq, 0, stream>>>(s, dout);
}